// SampleAndAggregate_25752623907093
// MI455X (gfx1250) — compile-verified
//
#include <hip/hip_runtime.h>
#include <hip/hip_bf16.h>

typedef __attribute__((ext_vector_type(2))) float v2f;
typedef __attribute__((ext_vector_type(8))) float v8f;

#define FEAT 256
#define TILE_K 32

// ---------------------------------------------------------------------------
// s1[i] = adj[batch[i/10]*25 + (i%10)]   for i in [0, 5120)
// ---------------------------------------------------------------------------
__global__ __launch_bounds__(256) void build_s1_kernel(
    const int* __restrict__ adj, const int* __restrict__ batch,
    int* __restrict__ s1)
{
    int i = blockIdx.x * 256 + threadIdx.x;
    if (i < 512 * 10) {
        int b = i / 10;
        int j = i - b * 10;
        s1[i] = adj[(size_t)batch[b] * 25 + j];
    }
}

// ---------------------------------------------------------------------------
// out[row] = mean over k<25 of features[adj[s1[row]*25+k]]   (row < 5120)
// one block (64 threads) per output row, float4 per thread (64*4 = 256 cols)
// ---------------------------------------------------------------------------
__global__ __launch_bounds__(64) void mean_neigh25_kernel(
    const float* __restrict__ feat, const int* __restrict__ adj,
    const int* __restrict__ s1, float* __restrict__ out)
{
    const int row  = blockIdx.x;
    const int t    = threadIdx.x;
    const int node = s1[row];                       // uniform -> SGPR
    const int* arow = adj + (size_t)node * 25;
    float ax = 0.f, ay = 0.f, az = 0.f, aw = 0.f;
    for (int k = 0; k < 25; ++k) {
        const int nb = arow[k];                     // uniform -> SGPR
        const float4 v = *(const float4*)(feat + (size_t)nb * FEAT + t * 4);
        ax += v.x; ay += v.y; az += v.z; aw += v.w;
    }
    const float s = 1.0f / 25.0f;
    float4 r; r.x = ax * s; r.y = ay * s; r.z = az * s; r.w = aw * s;
    *(float4*)(out + (size_t)row * FEAT + t * 4) = r;
}

// ---------------------------------------------------------------------------
// out[row] = mean over j<10 of features[s1[row*10+j]]   (row < 512)
// ---------------------------------------------------------------------------
__global__ __launch_bounds__(64) void mean_neigh10_kernel(
    const float* __restrict__ feat, const int* __restrict__ s1,
    float* __restrict__ out)
{
    const int row = blockIdx.x;
    const int t   = threadIdx.x;
    float ax = 0.f, ay = 0.f, az = 0.f, aw = 0.f;
    for (int j = 0; j < 10; ++j) {
        const int nb = s1[row * 10 + j];            // uniform -> SGPR
        const float4 v = *(const float4*)(feat + (size_t)nb * FEAT + t * 4);
        ax += v.x; ay += v.y; az += v.z; aw += v.w;
    }
    const float s = 0.1f;
    float4 r; r.x = ax * s; r.y = ay * s; r.z = az * s; r.w = aw * s;
    *(float4*)(out + (size_t)row * FEAT + t * 4) = r;
}

// ---------------------------------------------------------------------------
// out[row] = mean over j<10 of h1[row*10+j]  (contiguous rows, row < 512)
// ---------------------------------------------------------------------------
__global__ __launch_bounds__(64) void mean_rows10_kernel(
    const float* __restrict__ h1, float* __restrict__ out)
{
    const int row = blockIdx.x;
    const int t   = threadIdx.x;
    float ax = 0.f, ay = 0.f, az = 0.f, aw = 0.f;
    for (int j = 0; j < 10; ++j) {
        const float4 v =
            *(const float4*)(h1 + ((size_t)row * 10 + j) * FEAT + t * 4);
        ax += v.x; ay += v.y; az += v.z; aw += v.w;
    }
    const float s = 0.1f;
    float4 r; r.x = ax * s; r.y = ay * s; r.z = az * s; r.w = aw * s;
    *(float4*)(out + (size_t)row * FEAT + t * 4) = r;
}

// ---------------------------------------------------------------------------
// Fused GraphSAGE layer:  out = act(A_self @ Ws + A_neigh @ Wn)
//   A_self row m is features[idx[m]] when GATHER, else A_self[m]
//   A_neigh is dense [M x 256];  Ws/Wn are [256 x 256] row-major
//   M must be a multiple of 64.  N = K = 256.
// 256 threads = 8 waves; wave (mt,nh) computes rows [mt*16,+16) x cols
// [nh*128,+128) of the block's 64x256 output using f32 WMMA 16x16x4.
//
// Weights staged through LDS in K-chunks of 32 in a fragment-interleaved
// layout:  lds[kpair*512 + n*2 + (k&1)] = W[k0 + 2*kpair + (k&1)][n]
// so each B fragment {W[k][n], W[k+1][n]} is one aligned ds_load_b64
// directly into the WMMA source VGPR pair.  Each k-step first loads all
// 16 B fragments (clause of ds_load_2addr_b64, one s_wait), then issues
// 16 independent back-to-back WMMAs.
// ---------------------------------------------------------------------------
template <bool RELU, bool GATHER>
__global__ __launch_bounds__(256) void sage_gemm_kernel(
    const float* __restrict__ Aself, const int* __restrict__ idx,
    const float* __restrict__ Aneigh,
    const float* __restrict__ Ws, const float* __restrict__ Wn,
    float* __restrict__ out)
{
    __shared__ float lds_ws[TILE_K * FEAT];
    __shared__ float lds_wn[TILE_K * FEAT];

    const int tid       = threadIdx.x;
    const int lane      = tid & 31;
    const int wave      = tid >> 5;
    const int mtile     = wave >> 1;            // 0..3
    const int nhalf     = wave & 1;             // 0..1
    const int rowInTile = lane & 15;            // A: M index, B/C/D: N index
    const int klane     = (lane >> 4) << 1;     // 0 or 2 (K sub-pair)
    const int m         = blockIdx.x * 64 + mtile * 16 + rowInTile;

    const float* arow_s =
        GATHER ? (Aself + (size_t)idx[m] * FEAT) : (Aself + (size_t)m * FEAT);
    const float* arow_n = Aneigh + (size_t)m * FEAT;

    // per-lane base offset into the interleaved LDS chunk (floats)
    const int nBase = (nhalf * 128 + rowInTile) * 2;

    v8f acc[8];
#pragma unroll
    for (int t = 0; t < 8; ++t) acc[t] = (v8f)0.f;

    for (int k0 = 0; k0 < FEAT; k0 += TILE_K) {
        __syncthreads();  // previous chunk's LDS reads done
        {
            // stage one TILE_K x 256 chunk of each weight matrix,
            // interleaving k-pairs: one b128 LDS store covers
            // {W[2p][n], W[2p+1][n], W[2p][n+1], W[2p+1][n+1]}
            const float* gws = Ws + (size_t)k0 * FEAT;
            const float* gwn = Wn + (size_t)k0 * FEAT;
#pragma unroll
            for (int i = 0; i < 8; ++i) {
                const int u     = i * 256 + tid;   // 0..2047 pack units
                const int kpair = u >> 7;          // 0..15
                const int npair = u & 127;         // 0..127
                const int g0 = (2 * kpair) * FEAT + 2 * npair;
                const int g1 = g0 + FEAT;
                const int lo = kpair * 512 + npair * 4;
                {
                    const v2f r0 = *(const v2f*)(gws + g0);
                    const v2f r1 = *(const v2f*)(gws + g1);
                    float4 p; p.x = r0.x; p.y = r1.x; p.z = r0.y; p.w = r1.y;
                    *(float4*)(lds_ws + lo) = p;
                }
                {
                    const v2f r0 = *(const v2f*)(gwn + g0);
                    const v2f r1 = *(const v2f*)(gwn + g1);
                    float4 p; p.x = r0.x; p.y = r1.x; p.z = r0.y; p.w = r1.y;
                    *(float4*)(lds_wn + lo) = p;
                }
            }
        }
        __syncthreads();  // LDS populated

#pragma unroll
        for (int kk = 0; kk < TILE_K; kk += 4) {
            // A fragments (16x4 f32): lane holds K = klane, klane+1
            const v2f a_s = *(const v2f*)(arow_s + k0 + kk + klane);
            const v2f a_n = *(const v2f*)(arow_n + k0 + kk + klane);
            const int kp  = (kk + klane) >> 1;     // k-pair within chunk

            // phase 1: fetch all 16 B fragments (distinct regs -> one clause)
            v2f bs[8], bn[8];
#pragma unroll
            for (int t = 0; t < 8; ++t) {
                bs[t] = *(const v2f*)(lds_ws + kp * 512 + nBase + t * 32);
                bn[t] = *(const v2f*)(lds_wn + kp * 512 + nBase + t * 32);
            }
            // phase 2: 16 independent WMMAs, back-to-back
#pragma unroll
            for (int t = 0; t < 8; ++t) {
                acc[t] = __builtin_amdgcn_wmma_f32_16x16x4_f32(
                    false, a_s, false, bs[t], (short)0, acc[t], false, false);
                acc[t] = __builtin_amdgcn_wmma_f32_16x16x4_f32(
                    false, a_n, false, bn[t], (short)0, acc[t], false, false);
            }
        }
    }

    // write back: acc[t] vgpr r -> row = base + r + 8*(lane>>4), col = n
    const int colBase = nhalf * 128 + rowInTile;
    const int rowBase = blockIdx.x * 64 + mtile * 16 + ((lane >> 4) << 3);
#pragma unroll
    for (int t = 0; t < 8; ++t) {
#pragma unroll
        for (int r = 0; r < 8; ++r) {
            float v = acc[t][r];
            if (RELU) v = fmaxf(v, 0.0f);
            out[(size_t)(rowBase + r) * FEAT + colBase + t * 16] = v;
        }
    }
}

// ---------------------------------------------------------------------------
// L2-normalize each 256-wide row:  out = row / max(||row||, 1e-12)
// ---------------------------------------------------------------------------
__global__ __launch_bounds__(256) void l2norm_kernel(
    const float* __restrict__ pre, float* __restrict__ out)
{
    __shared__ float ssum[8];
    const int row = blockIdx.x;
    const int t   = threadIdx.x;
    const float v = pre[(size_t)row * FEAT + t];
    float s = v * v;
#pragma unroll
    for (int o = 16; o >= 1; o >>= 1) s += __shfl_xor(s, o, 32);
    if ((t & 31) == 0) ssum[t >> 5] = s;
    __syncthreads();
    float tot = 0.f;
#pragma unroll
    for (int i = 0; i < 8; ++i) tot += ssum[i];
    const float norm = fmaxf(sqrtf(tot), 1e-12f);
    out[(size_t)row * FEAT + t] = v / norm;
}

// ---------------------------------------------------------------------------
extern "C" void kernel_launch(void* const* d_in, const int* in_sizes, int n_in,
                              void* d_out, int out_size, void* d_ws,
                              size_t ws_size, hipStream_t stream)
{
    const float* features = (const float*)d_in[0];   // [100000 x 256]
    const int*   adj      = (const int*)d_in[1];     // [100000 x 25]
    const int*   batch    = (const int*)d_in[2];     // [512]
    const float* Ws0      = (const float*)d_in[3];   // [256 x 256]
    const float* Wn0      = (const float*)d_in[4];
    const float* Ws1      = (const float*)d_in[5];
    const float* Wn1      = (const float*)d_in[6];
    float* out = (float*)d_out;                      // [512 x 256]

    char* ws = (char*)d_ws;
    int*   s1   = (int*)(ws + 0);               //  5120 ints
    float* nm2  = (float*)(ws + 20480);         //  5120 x 256
    float* nm1  = (float*)(ws + 5263360);       //   512 x 256
    float* h1   = (float*)(ws + 5787648);       //  5120 x 256
    float* h0   = (float*)(ws + 11030528);      //   512 x 256
    float* nmh1 = (float*)(ws + 11554816);      //   512 x 256
    float* pre  = (float*)(ws + 12079104);      //   512 x 256

    // hop-1 sampled indices
    build_s1_kernel<<<(5120 + 255) / 256, 256, 0, stream>>>(adj, batch, s1);
    // fused gather + mean aggregations (hop-2 and hop-1 neighborhoods)
    mean_neigh25_kernel<<<5120, 64, 0, stream>>>(features, adj, s1, nm2);
    mean_neigh10_kernel<<<512, 64, 0, stream>>>(features, s1, nm1);
    // layer 0 (relu): hop-1 then hop-0
    sage_gemm_kernel<true, true><<<5120 / 64, 256, 0, stream>>>(
        features, s1, nm2, Ws0, Wn0, h1);
    sage_gemm_kernel<true, true><<<512 / 64, 256, 0, stream>>>(
        features, batch, nm1, Ws0, Wn0, h0);
    // layer 1: mean of h1 groups of 10, then final (no activation)
    mean_rows10_kernel<<<512, 64, 0, stream>>>(h1, nmh1);
    sage_gemm_kernel<false, false><<<512 / 64, 256, 0, stream>>>(
        h0, (const int*)nullptr, nmh1, Ws1, Wn1, pre);
    // l2 normalize rows
    l2norm_kernel<<<512, 256, 0, stream>>>(pre, out);
}